// ContactsFittingLoss_39101382263200
// MI455X (gfx1250) — compile-verified
//
#include <hip/hip_runtime.h>
#include <math.h>

#define N_ANCHORS 32
#define GP_STRIDE 14   // mean(3) + inv(9) + logdet(1) + zeroflag(1)
#define LOG_2PI 1.8378770664093453f
#define CHUNK 32       // obj tiles staged per round: 32*16 pts = 6 KB/buffer

typedef float v2f __attribute__((ext_vector_type(2)));
typedef float v8f __attribute__((ext_vector_type(8)));

// ---------------------------------------------------------------------------
// Kernel A: per-vertex sum of the 5 smallest squared distances to obj points.
// V_WMMA_F32_16X16X4_F32 computes 16x16 d2 tiles; obj points are streamed
// through a double-buffered LDS pipeline shared by all 8 waves of the block.
// ---------------------------------------------------------------------------
__global__ __launch_bounds__(256)
void knn5_wmma_kernel(const float* __restrict__ verts,
                      const float* __restrict__ obj,
                      float* __restrict__ knnsum,
                      int N, int P)
{
    __shared__ float sobj[2][CHUNK * 16 * 3];   // 2 x 6 KB

    const int lane = threadIdx.x & 31;
    const int wave = threadIdx.x >> 5;
    const int tile = blockIdx.x * 8 + wave;     // 8 vert-tiles per block
    const int vbase = tile * 16;
    const int idx  = lane & 15;                 // column (vert) / row (obj)
    const int half = lane >> 4;                 // K-split half per ISA layout

    // Vert (B-matrix) operand: rows K = (vx, vy, vz, 1); hoisted.
    const int vclamp = min(vbase + idx, N - 1);
    const float* vp = verts + (size_t)vclamp * 3;
    const float vx = vp[0], vy = vp[1], vz = vp[2];
    const float vn2 = vx * vx + vy * vy + vz * vz;
    v2f Bm;
    Bm[0] = half ? vz   : vx;   // VGPR0: K=0 (lanes 0-15) / K=2 (lanes 16-31)
    Bm[1] = half ? 1.0f : vy;   // VGPR1: K=1 / K=3

    // Sorted 5-smallest list (t0 <= ... <= t4) of squared distances.
    float t0 = 3.0e38f, t1 = 3.0e38f, t2 = 3.0e38f, t3 = 3.0e38f, t4 = 3.0e38f;

    const int ntiles  = P >> 4;
    const int nchunks = ntiles / CHUNK;

    // Prime buffer 0 with chunk 0.
    if (nchunks > 0) {
#pragma unroll
        for (int k = 0; k < 6; ++k)
            sobj[0][threadIdx.x + k * 256] = obj[threadIdx.x + k * 256];
    }
    __syncthreads();

    for (int c = 0; c < nchunks; ++c) {
        const int cur = c & 1, nxt = cur ^ 1;

        // Issue next chunk's global loads into registers; the waits land
        // after this chunk's compute, hiding the L2 latency.
        float stage[6];
        const bool have_next = (c + 1 < nchunks);
        if (have_next) {
            const float* src = obj + (size_t)(c + 1) * (CHUNK * 16 * 3);
#pragma unroll
            for (int k = 0; k < 6; ++k)
                stage[k] = src[threadIdx.x + k * 256];
        }

        // Process CHUNK obj tiles out of LDS (conflict-free, broadcast halves).
        for (int tt = 0; tt < CHUNK; ++tt) {
            const int ob = (tt * 16 + idx) * 3;
            const float ox = sobj[cur][ob + 0];
            const float oy = sobj[cur][ob + 1];
            const float oz = sobj[cur][ob + 2];
            const float on2 = ox * ox + oy * oy + oz * oz;
            v2f Am;                       // A row m=idx: (-2ox,-2oy,-2oz,|o|^2)
            Am[0] = half ? (-2.0f * oz) : (-2.0f * ox);
            Am[1] = half ? on2          : (-2.0f * oy);

            v8f acc = {};
            acc = __builtin_amdgcn_wmma_f32_16x16x4_f32(
                false, Am, false, Bm, (short)0, acc, false, false);

            // acc[r] = -2*o.v + |o|^2 for obj row (r + 8*half); +|v|^2 -> d2
            float d[8];
#pragma unroll
            for (int r = 0; r < 8; ++r)
                d[r] = fmaxf(acc[r] + vn2, 0.0f);
            const float m = fminf(fminf(fminf(d[0], d[1]), fminf(d[2], d[3])),
                                  fminf(fminf(d[4], d[5]), fminf(d[6], d[7])));
            if (__ballot(m < t4)) {       // rare once the list warms up
#pragma unroll
                for (int r = 0; r < 8; ++r) {
                    const float v = d[r];
                    const bool c0 = v < t0, c1 = v < t1, c2 = v < t2,
                               c3 = v < t3, c4 = v < t4;
                    t4 = c4 ? (c3 ? t3 : v) : t4;
                    t3 = c3 ? (c2 ? t2 : v) : t3;
                    t2 = c2 ? (c1 ? t1 : v) : t2;
                    t1 = c1 ? (c0 ? t0 : v) : t1;
                    t0 = c0 ? v : t0;
                }
            }
        }

        // Commit staged registers to the other buffer, then rotate.
        if (have_next) {
#pragma unroll
            for (int k = 0; k < 6; ++k)
                sobj[nxt][threadIdx.x + k * 256] = stage[k];
        }
        __syncthreads();
    }

    // Remainder tiles (P not a multiple of CHUNK*16): straight global path.
    for (int t = nchunks * CHUNK; t < ntiles; ++t) {
        const float* op = obj + (size_t)(t * 16 + idx) * 3;
        const float ox = op[0], oy = op[1], oz = op[2];
        const float on2 = ox * ox + oy * oy + oz * oz;
        v2f Am;
        Am[0] = half ? (-2.0f * oz) : (-2.0f * ox);
        Am[1] = half ? on2          : (-2.0f * oy);
        v8f acc = {};
        acc = __builtin_amdgcn_wmma_f32_16x16x4_f32(
            false, Am, false, Bm, (short)0, acc, false, false);
#pragma unroll
        for (int r = 0; r < 8; ++r) {
            const float v = fmaxf(acc[r] + vn2, 0.0f);
            const bool c0 = v < t0, c1 = v < t1, c2 = v < t2,
                       c3 = v < t3, c4 = v < t4;
            t4 = c4 ? (c3 ? t3 : v) : t4;
            t3 = c3 ? (c2 ? t2 : v) : t3;
            t2 = c2 ? (c1 ? t1 : v) : t2;
            t1 = c1 ? (c0 ? t0 : v) : t1;
            t0 = c0 ? v : t0;
        }
    }

    // Merge with partner half (lane ^ 16): snapshot first, then insert.
    const float p0 = __shfl_xor(t0, 16, 32);
    const float p1 = __shfl_xor(t1, 16, 32);
    const float p2 = __shfl_xor(t2, 16, 32);
    const float p3 = __shfl_xor(t3, 16, 32);
    const float p4 = __shfl_xor(t4, 16, 32);
    const float pv[5] = {p0, p1, p2, p3, p4};
#pragma unroll
    for (int j = 0; j < 5; ++j) {
        const float v = pv[j];
        const bool c0 = v < t0, c1 = v < t1, c2 = v < t2, c3 = v < t3, c4 = v < t4;
        t4 = c4 ? (c3 ? t3 : v) : t4;
        t3 = c3 ? (c2 ? t2 : v) : t3;
        t2 = c2 ? (c1 ? t1 : v) : t2;
        t1 = c1 ? (c0 ? t0 : v) : t1;
        t0 = c0 ? v : t0;
    }

    if (half == 0 && (vbase + idx) < N)
        knnsum[vbase + idx] = t0 + t1 + t2 + t3 + t4;  // = sum_k knn_k^2
}

// ---------------------------------------------------------------------------
// Kernel B: per-anchor gaussian prep (3x3 inverse, logdet, zero flag) + inits.
// ---------------------------------------------------------------------------
__global__ void gauss_prep_kernel(const float* __restrict__ cg,
                                  const float* __restrict__ anchors,
                                  float* __restrict__ gp,
                                  unsigned int* __restrict__ gmax,
                                  float* __restrict__ out)
{
    const int k = threadIdx.x;
    if (k == 0) out[0] = 0.0f;
    if (k >= N_ANCHORS) return;
    gmax[k] = 0u;  // exp(logp) > 0, so uint bit-order == float order

    const float* g = cg + k * 12;
    bool allz = true;
#pragma unroll
    for (int i = 0; i < 12; ++i) allz &= (g[i] == 0.0f);

    float c00 = g[3], c01 = g[4],  c02 = g[5];
    float c10 = g[6], c11 = g[7],  c12 = g[8];
    float c20 = g[9], c21 = g[10], c22 = g[11];
    if (allz) {
        c00 = 1.f; c01 = 0.f; c02 = 0.f;
        c10 = 0.f; c11 = 1.f; c12 = 0.f;
        c20 = 0.f; c21 = 0.f; c22 = 1.f;
    }
    const float det = c00 * (c11 * c22 - c12 * c21)
                    - c01 * (c10 * c22 - c12 * c20)
                    + c02 * (c10 * c21 - c11 * c20);
    const float id = 1.0f / det;

    float* o = gp + k * GP_STRIDE;
    o[0] = g[0] + anchors[k * 3 + 0];
    o[1] = g[1] + anchors[k * 3 + 1];
    o[2] = g[2] + anchors[k * 3 + 2];
    o[3]  =  (c11 * c22 - c12 * c21) * id;
    o[4]  = -(c01 * c22 - c02 * c21) * id;
    o[5]  =  (c01 * c12 - c02 * c11) * id;
    o[6]  = -(c10 * c22 - c12 * c20) * id;
    o[7]  =  (c00 * c22 - c02 * c20) * id;
    o[8]  = -(c00 * c12 - c02 * c10) * id;
    o[9]  =  (c10 * c21 - c11 * c20) * id;
    o[10] = -(c00 * c21 - c01 * c20) * id;
    o[11] =  (c00 * c11 - c01 * c10) * id;
    o[12] = logf(det);               // SPD: 2*logdet(chol) == log(det)
    o[13] = allz ? 1.0f : 0.0f;
}

// ---------------------------------------------------------------------------
// Kernel C: nearest anchor, Mahalanobis weight, group-max via atomicMax.
// ---------------------------------------------------------------------------
__global__ __launch_bounds__(256)
void weights_kernel(const float* __restrict__ verts,
                    const float* __restrict__ anchors,
                    const float* __restrict__ gp,
                    float* __restrict__ w,
                    int* __restrict__ aidx,
                    unsigned int* __restrict__ gmax,
                    int N)
{
    const int i = blockIdx.x * 256 + threadIdx.x;
    if (i >= N) return;
    const float vx = verts[i * 3 + 0], vy = verts[i * 3 + 1], vz = verts[i * 3 + 2];

    int best = 0;
    float bd = 3.0e38f;
#pragma unroll
    for (int k = 0; k < N_ANCHORS; ++k) {
        const float dx = vx - anchors[k * 3 + 0];
        const float dy = vy - anchors[k * 3 + 1];
        const float dz = vz - anchors[k * 3 + 2];
        const float d2 = dx * dx + dy * dy + dz * dz;
        if (d2 < bd) { bd = d2; best = k; }   // strict < keeps first (argmin)
    }

    const float* o = gp + best * GP_STRIDE;
    const float dx = vx - o[0], dy = vy - o[1], dz = vz - o[2];
    const float maha = dx * (o[3] * dx + o[4]  * dy + o[5]  * dz)
                     + dy * (o[6] * dx + o[7]  * dy + o[8]  * dz)
                     + dz * (o[9] * dx + o[10] * dy + o[11] * dz);
    const float logp = -0.5f * (maha + o[12] + 3.0f * LOG_2PI);
    const float wv = expf(logp);

    w[i] = wv;
    aidx[i] = best;
    atomicMax(&gmax[best], __float_as_uint(wv));
}

// ---------------------------------------------------------------------------
// Kernel D: normalize/threshold/mask weights, fold with knn^2 sums, reduce.
// ---------------------------------------------------------------------------
__global__ __launch_bounds__(256)
void loss_kernel(const float* __restrict__ w,
                 const int* __restrict__ aidx,
                 const unsigned int* __restrict__ gmax,
                 const float* __restrict__ gp,
                 const float* __restrict__ knnsum,
                 const int* __restrict__ Kptr,
                 float* __restrict__ out,
                 int N)
{
    const int i = blockIdx.x * 256 + threadIdx.x;
    float term = 0.0f;
    if (i < N) {
        const int a = aidx[i];
        const float g = __uint_as_float(gmax[a]);
        const float norm = (g > 1.0f) ? g : 1.0f;
        float wv = w[i] / norm;
        wv = (wv > 0.01f) ? wv : 0.0f;
        if (gp[a * GP_STRIDE + 13] != 0.0f) wv = 0.0f;  // all-zero gaussian
        term = wv * wv * knnsum[i];
    }
    // wave32 tree reduce, then cross-wave via LDS, then one atomic per block
#pragma unroll
    for (int off = 16; off > 0; off >>= 1)
        term += __shfl_down(term, off, 32);

    __shared__ float red[8];
    const int lane = threadIdx.x & 31, wid = threadIdx.x >> 5;
    if (lane == 0) red[wid] = term;
    __syncthreads();
    if (wid == 0) {
        float v = (lane < 8) ? red[lane] : 0.0f;
#pragma unroll
        for (int off = 4; off > 0; off >>= 1)
            v += __shfl_down(v, off, 32);
        if (lane == 0) {
            const float scale = 1.0f / ((float)N * (float)Kptr[0]);
            atomicAdd(out, v * scale);
        }
    }
}

// ---------------------------------------------------------------------------
extern "C" void kernel_launch(void* const* d_in, const int* in_sizes, int n_in,
                              void* d_out, int out_size, void* d_ws, size_t ws_size,
                              hipStream_t stream)
{
    const float* verts   = (const float*)d_in[0];
    const float* anchors = (const float*)d_in[1];
    const float* obj     = (const float*)d_in[2];
    const float* cg      = (const float*)d_in[3];
    const int*   Kp      = (const int*)d_in[4];
    const int N = in_sizes[0] / 3;
    const int P = in_sizes[2] / 3;

    float* ws = (float*)d_ws;
    float*        knnsum = ws;                                       // N floats
    float*        gp     = ws + N;                                   // 32*14
    unsigned int* gmax   = (unsigned int*)(ws + N + N_ANCHORS * GP_STRIDE);  // 32
    float*        w      = ws + N + N_ANCHORS * GP_STRIDE + N_ANCHORS;       // N
    int*          aidx   = (int*)(w + N);                            // N ints
    float* out = (float*)d_out;

    const int tiles = (N + 15) / 16;             // one wave per 16 verts
    knn5_wmma_kernel<<<dim3((tiles + 7) / 8), dim3(256), 0, stream>>>(
        verts, obj, knnsum, N, P);
    gauss_prep_kernel<<<dim3(1), dim3(32), 0, stream>>>(cg, anchors, gp, gmax, out);
    const int blocks = (N + 255) / 256;
    weights_kernel<<<dim3(blocks), dim3(256), 0, stream>>>(
        verts, anchors, gp, w, aidx, gmax, N);
    loss_kernel<<<dim3(blocks), dim3(256), 0, stream>>>(
        w, aidx, gmax, gp, knnsum, Kp, out, N);
}